// PixPro_41678362640819
// MI455X (gfx1250) — compile-verified
//
#include <hip/hip_runtime.h>

// ---------------------------------------------------------------------------
// Factorized PixPro loss for MI455X (gfx1250, wave32).
//
//   num[n] = sum_{k=1..7} < sum_{p: rm=k} rgb_hat[:,p] , sum_{q: im=k} ir_hat[:,q] >
//   den[n] = sum_{k=1..7} cnt_rgb[k]*cnt_ir[k] + 1e-6
//   out    = -mean_n num/den
//
// The clip(-1,1) in the reference is inert for normalized random vectors
// (|dot| <~ 0.5), so the factorization is exact to fp rounding.
//
// Label-sum S[n][src][k][c] = sum_p (label_p==k)*inv_p * X[c][p] is computed
// as a 16x128x4096 GEMM per (n,src) with V_WMMA_F32_16X16X4_F32:
//   A[m][p] = (label_p==m) ? inv_p : 0      (built in registers)
//   B[p][c] = raw map value                  (normalization folded into A)
// ---------------------------------------------------------------------------

typedef float v2f __attribute__((ext_vector_type(2)));
typedef float v8f __attribute__((ext_vector_type(8)));

#define NB   8
#define CCH  128
#define PP   4096
#define KSEG 4          // K (pixel) segments per GEMM -> 512 one-wave blocks

// ---- Kernel 1: per-pixel inverse L2 norms (coalesced over p) --------------
__global__ void __launch_bounds__(256) k_norms(const float* __restrict__ rgb,
                                               const float* __restrict__ ir,
                                               float* __restrict__ invr,
                                               float* __restrict__ invi) {
  int idx = blockIdx.x * 256 + threadIdx.x;   // n*4096 + p
  int n = idx >> 12;
  int p = idx & (PP - 1);
  const float* r = rgb + (size_t)n * CCH * PP + p;
  const float* i = ir  + (size_t)n * CCH * PP + p;
  float sr = 0.f, si = 0.f;
#pragma unroll 8
  for (int c = 0; c < CCH; ++c) {
    float a = r[(size_t)c * PP]; sr += a * a;
    float b = i[(size_t)c * PP]; si += b * b;
  }
  invr[idx] = 1.0f / fmaxf(sqrtf(sr), 1e-12f);   // matches F.normalize eps
  invi[idx] = 1.0f / fmaxf(sqrtf(si), 1e-12f);
}

// ---- Kernel 2: label-indicator GEMM via V_WMMA_F32_16X16X4_F32 ------------
// One wave per block. block b: kseg = b&3, src = (b>>2)&1, ctile = (b>>3)&7,
// n = b>>6.  Output: partial S, layout [n][src][kseg][m=16][c=128].
__global__ void __launch_bounds__(32) k_label_gemm(const float* __restrict__ rgb,
                                                   const float* __restrict__ ir,
                                                   const int*  __restrict__ rmask,
                                                   const int*  __restrict__ imask,
                                                   const float* __restrict__ invr,
                                                   const float* __restrict__ invi,
                                                   float* __restrict__ Sp) {
  const int b    = blockIdx.x;
  const int kseg = b & (KSEG - 1);
  const int src  = (b >> 2) & 1;
  const int ct   = (b >> 3) & 7;
  const int n    = b >> 6;

  const int lane = threadIdx.x;       // 0..31
  const int m    = lane & 15;         // A row (label) == B col (channel) index
  const int half = lane >> 4;         // 0: K={0,1}, 1: K={2,3} within a step

  const float* X   = (src ? ir    : rgb)  + (size_t)n * CCH * PP;
  const int*   lab = (src ? imask : rmask) + (size_t)n * PP;
  const float* inv = (src ? invi  : invr) + (size_t)n * PP;

  const int c0 = ct * 16;
  const float* Xrow = X + (size_t)(c0 + m) * PP;   // this lane's channel row
  const int pbase = kseg * (PP / KSEG) + 2 * half;

  v8f acc0 = {0.f, 0.f, 0.f, 0.f, 0.f, 0.f, 0.f, 0.f};
  v8f acc1 = acc0;

  // 1024/KSEG = 256 K-steps of 4 pixels each; two accumulators hide the
  // WMMA->WMMA RAW accumulation chain.
  for (int s = 0; s < (PP / KSEG) / 4; s += 2) {
    const int p0 = pbase + 4 * s;
    const int p1 = p0 + 4;

    int2   l0 = *(const int2*)  (lab  + p0);
    float2 v0 = *(const float2*)(inv  + p0);
    float2 x0 = *(const float2*)(Xrow + p0);
    v2f a0, b0;
    a0.x = (l0.x == m) ? v0.x : 0.0f;
    a0.y = (l0.y == m) ? v0.y : 0.0f;
    b0.x = x0.x; b0.y = x0.y;
    acc0 = __builtin_amdgcn_wmma_f32_16x16x4_f32(false, a0, false, b0,
                                                 (short)0, acc0, false, false);

    int2   l1 = *(const int2*)  (lab  + p1);
    float2 v1 = *(const float2*)(inv  + p1);
    float2 x1 = *(const float2*)(Xrow + p1);
    v2f a1, b1;
    a1.x = (l1.x == m) ? v1.x : 0.0f;
    a1.y = (l1.y == m) ? v1.y : 0.0f;
    b1.x = x1.x; b1.y = x1.y;
    acc1 = __builtin_amdgcn_wmma_f32_16x16x4_f32(false, a1, false, b1,
                                                 (short)0, acc1, false, false);
  }

  v8f d = acc0 + acc1;

  // D layout: VGPR j -> M = j + 8*(lane>=16), N = lane%16.
  float* out = Sp + ((((size_t)n * 2 + src) * KSEG + kseg) * 16) * CCH;
#pragma unroll
  for (int j = 0; j < 8; ++j) {
    int row = j + 8 * half;
    out[(size_t)row * CCH + c0 + m] = d[j];
  }
}

// ---- Kernel 3: histograms + trace + final scalar (single block, det.) -----
__global__ void __launch_bounds__(256) k_finalize(const int* __restrict__ rmask,
                                                  const int* __restrict__ imask,
                                                  const float* __restrict__ Sp,
                                                  float* __restrict__ out) {
  __shared__ int   cnt[2][NB][8];
  __shared__ float lpart[NB];
  const int t = threadIdx.x;

  if (t < 128) {                       // 128 tasks: (n, src, k) histograms
    const int k   = t & 7;
    const int src = (t >> 3) & 1;
    const int n   = t >> 4;
    const int* lab = (src ? imask : rmask) + (size_t)n * PP;
    int c = 0;
    for (int p = 0; p < PP; ++p) c += (lab[p] == k);
    cnt[src][n][k] = c;
  }
  __syncthreads();

  if (t < NB) {                        // one thread per batch: num/den
    const int n = t;
    float num = 0.f;
    const float* sr = Sp + (((size_t)n * 2 + 0) * KSEG) * 16 * CCH;
    const float* si = Sp + (((size_t)n * 2 + 1) * KSEG) * 16 * CCH;
    for (int k = 1; k < 8; ++k) {
      for (int c = 0; c < CCH; ++c) {
        const int o = k * CCH + c;
        float a = sr[o] + sr[o + 2048] + sr[o + 4096] + sr[o + 6144];
        float b = si[o] + si[o + 2048] + si[o + 4096] + si[o + 6144];
        num += a * b;
      }
    }
    float den = 1e-6f;
    for (int k = 1; k < 8; ++k)
      den += (float)cnt[0][n][k] * (float)cnt[1][n][k];
    lpart[n] = num / den;
  }
  __syncthreads();

  if (t == 0) {
    float s = 0.f;
    for (int n = 0; n < NB; ++n) s += lpart[n];
    out[0] = -s * (1.0f / NB);
  }
}

extern "C" void kernel_launch(void* const* d_in, const int* in_sizes, int n_in,
                              void* d_out, int out_size, void* d_ws, size_t ws_size,
                              hipStream_t stream) {
  const float* rgb = (const float*)d_in[0];
  const float* ir  = (const float*)d_in[1];
  const int*   rm  = (const int*)d_in[2];
  const int*   im  = (const int*)d_in[3];

  float* ws   = (float*)d_ws;
  float* invr = ws;                       //  32768 floats
  float* invi = ws + NB * PP;             //  32768 floats
  float* Sp   = ws + 2 * NB * PP;         // 8*2*4*16*128 = 131072 floats
  // total workspace: 786 KB

  k_norms<<<NB * PP / 256, 256, 0, stream>>>(rgb, ir, invr, invi);
  k_label_gemm<<<NB * 2 * 8 * KSEG, 32, 0, stream>>>(rgb, ir, rm, im,
                                                     invr, invi, Sp);
  k_finalize<<<1, 256, 0, stream>>>(rm, im, Sp, (float*)d_out);
}